// NeuralDisCoCirc_45990509806199
// MI455X (gfx1250) — compile-verified
//
#include <hip/hip_runtime.h>

typedef __attribute__((ext_vector_type(2))) float v2f;
typedef __attribute__((ext_vector_type(8))) float v8f;

namespace {
constexpr int kL = 8;      // layers
constexpr int kB = 32;     // batch
constexpr int kD = 1280;   // width
constexpr int kWaves = 4;  // waves per block
constexpr int kBlock = kWaves * 32;  // 128 threads
}

// One layer: hOut[b,j] = maskedrelu( sum_i hIn[b,i]*W[b,i,j] + bias[b,j] )
// Each wave computes a 16-wide j tile of one batch element's GEMV using
// v_wmma_f32_16x16x4_f32 with the h chunk broadcast into all 16 M rows
// (row 0 of C carries the true result; other rows are identical copies).
__global__ __launch_bounds__(kBlock)
void ndc_layer_wmma(const float* __restrict__ hIn,   // [B, D]
                    const float* __restrict__ W,     // [B, D, D] (this layer)
                    const float* __restrict__ bias,  // [B, D]   (this layer)
                    const int*   __restrict__ mask,  // [B, D]   (this layer)
                    float* __restrict__ hOut)        // [B, D]
{
    __shared__ float sh[kD];

    const int b     = blockIdx.y;
    const int tid   = threadIdx.x;
    const int lane  = tid & 31;
    const int wave  = tid >> 5;
    const int n     = lane & 15;   // N index within the 16-wide j tile
    const int khalf = lane >> 4;   // 0: lanes 0-15 hold K=0,1 ; 1: lanes 16-31 hold K=2,3

    // Stage h[b,:] (5 KB) into LDS; uniform trip count keeps EXEC all-ones.
    const float* hb = hIn + b * kD;
    #pragma unroll
    for (int i = tid; i < kD; i += kBlock) sh[i] = hb[i];
    __syncthreads();

    const int j0 = (blockIdx.x * kWaves + wave) * 16;

    // B-fragment pointer for this lane: W[b, 2*khalf, j0+n]
    const float* wp = W + (size_t)b * kD * kD + (size_t)(2 * khalf) * kD + j0 + n;

    v8f c0 = {};
    v8f c1 = {};

    #pragma unroll 4
    for (int i = 0; i < kD; i += 8) {
        if ((i & 63) == 0) {
            // stream-ahead hint for the weight walk (global_prefetch_b8)
            __builtin_prefetch(wp + 64 * kD, 0, 1);
        }
        // ---- chain 0: rows i .. i+3 ----
        // A frag: all M rows = (h[i], h[i+1], h[i+2], h[i+3]);
        // lane-half selects K pair, so this is an LDS broadcast b64 load.
        v2f a0 = *(const v2f*)(&sh[i + 2 * khalf]);
        v2f b0;
        b0.x = wp[0];        // K = 2*khalf,   N = n
        b0.y = wp[kD];       // K = 2*khalf+1, N = n
        c0 = __builtin_amdgcn_wmma_f32_16x16x4_f32(
                 false, a0, false, b0, (short)0, c0, false, false);

        // ---- chain 1: rows i+4 .. i+7 (independent accumulator) ----
        v2f a1 = *(const v2f*)(&sh[i + 4 + 2 * khalf]);
        v2f b1;
        b1.x = wp[4 * kD];
        b1.y = wp[5 * kD];
        c1 = __builtin_amdgcn_wmma_f32_16x16x4_f32(
                 false, a1, false, b1, (short)0, c1, false, false);

        wp += 8 * kD;
    }

    // C/D layout: VGPR0, lanes 0-15 = (M=0, N=lane). Row 0 = true result.
    if (lane < 16) {
        const int j = j0 + n;
        float v = c0[0] + c1[0] + bias[b * kD + j];
        if (mask[b * kD + j] != 0) v = fmaxf(v, 0.0f);
        hOut[b * kD + j] = v;
    }
}

extern "C" void kernel_launch(void* const* d_in, const int* in_sizes, int n_in,
                              void* d_out, int out_size, void* d_ws, size_t ws_size,
                              hipStream_t stream) {
    (void)in_sizes; (void)n_in; (void)out_size; (void)ws_size;

    const float* x    = (const float*)d_in[0];  // [B, D]
    const float* W    = (const float*)d_in[1];  // [L, B, D, D]
    const float* bias = (const float*)d_in[2];  // [L, B, D]
    const int*   mask = (const int*)d_in[3];    // [L, B, D]
    float* out = (float*)d_out;                 // [B, D]
    float* tmp = (float*)d_ws;                  // one h buffer: B*D floats (160 KB)

    dim3 grid(kD / (16 * kWaves), kB);  // (80/kWaves) j-blocks x 32 batch = 640 blocks
    dim3 block(kBlock);

    // Ping-pong h between tmp (even layers) and d_out (odd layers);
    // kL is even-length so layer 7 writes d_out.
    for (int l = 0; l < kL; ++l) {
        const float* hin  = (l == 0) ? x : ((l & 1) ? tmp : out);
        float*       hout = (l & 1) ? out : tmp;
        hipLaunchKernelGGL(ndc_layer_wmma, grid, block, 0, stream,
                           hin,
                           W    + (size_t)l * kB * kD * kD,
                           bias + (size_t)l * kB * kD,
                           mask + (size_t)l * kB * kD,
                           hout);
    }
}